// DGC_VAE_13073880449913
// MI455X (gfx1250) — compile-verified
//
#include <hip/hip_runtime.h>
#include <hip/hip_bf16.h>

// Problem constants (match reference)
#define NN   16384
#define INC  256
#define HID  128
#define LAT  64
#define NE   262144

typedef float v2f __attribute__((ext_vector_type(2)));
typedef float v8f __attribute__((ext_vector_type(8)));

__device__ __forceinline__ v8f wmma4(v2f a, v2f b, v8f c) {
    // D = A(16x4 f32) x B(4x16 f32) + C(16x16 f32)
    return __builtin_amdgcn_wmma_f32_16x16x4_f32(
        /*neg_a=*/false, a, /*neg_b=*/false, b,
        /*c_mod=*/(short)0, c, /*reuse_a=*/false, /*reuse_b=*/false);
}

// ---------------------------------------------------------------------------
// Degree / norm kernels
// ---------------------------------------------------------------------------
__global__ void k_fill1(float* __restrict__ p, int n) {
    int i = blockIdx.x * blockDim.x + threadIdx.x;
    if (i < n) p[i] = 1.0f;   // self-loop contributes 1 to every node's degree
}

__global__ void k_deg_edges(const int* __restrict__ dst, float* __restrict__ deg) {
    int e = blockIdx.x * blockDim.x + threadIdx.x;
    if (e < NE) atomicAdd(&deg[dst[e]], 1.0f);
}

__global__ void k_rsqrt_inplace(float* __restrict__ p, int n) {
    int i = blockIdx.x * blockDim.x + threadIdx.x;
    if (i < n) {
        float d = p[i];
        p[i] = (d > 0.0f) ? rsqrtf(d) : 0.0f;
    }
}

// ---------------------------------------------------------------------------
// Generic fp32 WMMA GEMM: C[MxNc] = A[MxK] (row-major) * B[KxNc] (row-major)
// One wave computes one 16x16 output tile via v_wmma_f32_16x16x4_f32.
// ---------------------------------------------------------------------------
__global__ void k_gemm_f32(const float* __restrict__ A, const float* __restrict__ B,
                           float* __restrict__ C, int M, int K, int Nc) {
    const int wave   = blockIdx.x * (blockDim.x >> 5) + (threadIdx.x >> 5);
    const int tilesN = Nc >> 4;
    const int tm = wave / tilesN;
    const int tn = wave - tm * tilesN;
    if (tm * 16 >= M) return;                 // wave-uniform; EXEC stays all-1s
    const int lane = threadIdx.x & 31;
    const int half = lane >> 4;               // 0: K pair {0,1}, 1: K pair {2,3}
    const int l    = lane & 15;
    const int m0 = tm * 16, n0 = tn * 16;

    v8f acc = {};
    for (int k0 = 0; k0 < K; k0 += 4) {
        const int ka = k0 + 2 * half;
        // A fragment: row m0+l, contiguous pair of K values
        v2f a = *(const v2f*)(A + (long)(m0 + l) * K + ka);
        // B fragment: column n0+l, K values ka, ka+1 (row-major B -> strided)
        v2f b;
        b.x = B[(long)ka       * Nc + n0 + l];
        b.y = B[(long)(ka + 1) * Nc + n0 + l];
        acc = wmma4(a, b, acc);
    }
    // C/D layout: VGPR r -> row m0 + half*8 + r, col n0 + l
    const long col = n0 + l;
    const int  rb  = m0 + half * 8;
#pragma unroll
    for (int r = 0; r < 8; ++r)
        C[(long)(rb + r) * Nc + col] = acc[r];
}

// ---------------------------------------------------------------------------
// GCN message passing
// ---------------------------------------------------------------------------
// agg[i][c] = feat[i][c] * dinv[i]^2   (self-loop term; also initializes agg)
__global__ void k_selfinit(const float* __restrict__ feat, const float* __restrict__ dinv,
                           float* __restrict__ agg, int C, long total) {
    long i = (long)blockIdx.x * blockDim.x + threadIdx.x;
    if (i >= total) return;
    int node = (int)(i / C);
    float di = dinv[node];
    agg[i] = feat[i] * di * di;
}

// One wave per edge; 32 lanes stripe across C channels; atomic scatter to dst.
__global__ void k_edge_scatter(const float* __restrict__ feat, float* __restrict__ agg,
                               const int* __restrict__ src, const int* __restrict__ dst,
                               const float* __restrict__ dinv, int C) {
    const int e = blockIdx.x * (blockDim.x >> 5) + (threadIdx.x >> 5);
    if (e >= NE) return;
    const int lane = threadIdx.x & 31;
    const int s = src[e], d = dst[e];
    const float w = dinv[s] * dinv[d];
    for (int c = lane; c < C; c += 32)
        atomicAdd(&agg[(long)d * C + c], feat[(long)s * C + c] * w);
}

__global__ void k_bias_act(float* __restrict__ a, const float* __restrict__ b,
                           int C, long total, int relu) {
    long i = (long)blockIdx.x * blockDim.x + threadIdx.x;
    if (i >= total) return;
    float v = a[i] + b[(int)(i % C)];
    a[i] = relu ? fmaxf(v, 0.0f) : v;
}

// z = mu + eps * exp(0.5 * logvar)
__global__ void k_reparam(const float* __restrict__ mu, const float* __restrict__ lv,
                          const float* __restrict__ eps, float* __restrict__ z, long total) {
    long i = (long)blockIdx.x * blockDim.x + threadIdx.x;
    if (i >= total) return;
    z[i] = mu[i] + eps[i] * __expf(0.5f * lv[i]);
}

// ---------------------------------------------------------------------------
// recon = sigmoid(z @ z^T), z: [NN x LAT].
// One wave per 32x32 output tile: 2x2 blocking of 16x16 WMMA tiles, K=64.
// B(k, n) = z[n*LAT + k]  -> both A and B fragments are contiguous 8B loads.
// ---------------------------------------------------------------------------
__global__ void k_recon(const float* __restrict__ z, float* __restrict__ out) {
    const int wave  = blockIdx.x * (blockDim.x >> 5) + (threadIdx.x >> 5);
    const int tiles = NN >> 5;                 // 512 tiles per dimension
    const int tm = wave / tiles;
    const int tn = wave - tm * tiles;
    const int lane = threadIdx.x & 31;
    const int half = lane >> 4;
    const int l    = lane & 15;
    const int m0 = tm * 32, n0 = tn * 32;

    v8f c00 = {}, c01 = {}, c10 = {}, c11 = {};
    for (int k0 = 0; k0 < LAT; k0 += 4) {
        const int ka = k0 + 2 * half;
        v2f a0 = *(const v2f*)(z + (long)(m0      + l) * LAT + ka);
        v2f a1 = *(const v2f*)(z + (long)(m0 + 16 + l) * LAT + ka);
        v2f b0 = *(const v2f*)(z + (long)(n0      + l) * LAT + ka);
        v2f b1 = *(const v2f*)(z + (long)(n0 + 16 + l) * LAT + ka);
        c00 = wmma4(a0, b0, c00);
        c01 = wmma4(a0, b1, c01);
        c10 = wmma4(a1, b0, c10);
        c11 = wmma4(a1, b1, c11);
    }

    auto store16 = [&](int mb, int nb, v8f c) {
        const long col = nb + l;
        const int  rb  = mb + half * 8;
#pragma unroll
        for (int r = 0; r < 8; ++r) {
            float s = 1.0f / (1.0f + __expf(-c[r]));   // fused sigmoid
            out[(long)(rb + r) * NN + col] = s;
        }
    };
    store16(m0,      n0,      c00);
    store16(m0,      n0 + 16, c01);
    store16(m0 + 16, n0,      c10);
    store16(m0 + 16, n0 + 16, c11);
}

// ---------------------------------------------------------------------------
// Launch
// ---------------------------------------------------------------------------
extern "C" void kernel_launch(void* const* d_in, const int* in_sizes, int n_in,
                              void* d_out, int out_size, void* d_ws, size_t ws_size,
                              hipStream_t stream) {
    (void)in_sizes; (void)n_in; (void)out_size; (void)ws_size;

    const float* x   = (const float*)d_in[0];
    const int*   ei  = (const int*)  d_in[1];
    const float* eps = (const float*)d_in[2];
    const float* W1  = (const float*)d_in[3];
    const float* b1  = (const float*)d_in[4];
    const float* Wmu = (const float*)d_in[5];
    const float* bmu = (const float*)d_in[6];
    const float* Wlv = (const float*)d_in[7];
    const float* blv = (const float*)d_in[8];
    const int* srcI = ei;           // edge_index row 0
    const int* dstI = ei + NE;      // edge_index row 1

    float* out = (float*)d_out;
    float* mu  = out + (long)NN * NN;
    float* lv  = mu  + (long)NN * LAT;

    float* ws   = (float*)d_ws;
    float* dinv = ws;                         // NN
    float* xw   = dinv + NN;                  // NN*HID
    float* h    = xw   + (long)NN * HID;      // NN*HID
    float* hu   = h    + (long)NN * HID;      // NN*LAT
    float* hv   = hu   + (long)NN * LAT;      // NN*LAT
    float* zb   = hv   + (long)NN * LAT;      // NN*LAT

    const long nh = (long)NN * HID;           // 2,097,152
    const long nl = (long)NN * LAT;           // 1,048,576

    // 1) symmetric GCN norm: deg (with self loops) -> 1/sqrt(deg)
    k_fill1        <<<NN / 256, 256, 0, stream>>>(dinv, NN);
    k_deg_edges    <<<NE / 256, 256, 0, stream>>>(dstI, dinv);
    k_rsqrt_inplace<<<NN / 256, 256, 0, stream>>>(dinv, NN);

    // 2) layer 1: h = relu(scatter(norm * (x@W1)) + b1)
    k_gemm_f32    <<<(NN / 16) * (HID / 16) / 8, 256, 0, stream>>>(x, W1, xw, NN, INC, HID);
    k_selfinit    <<<nh / 256, 256, 0, stream>>>(xw, dinv, h, HID, nh);
    k_edge_scatter<<<NE / 8, 256, 0, stream>>>(xw, h, srcI, dstI, dinv, HID);
    k_bias_act    <<<nh / 256, 256, 0, stream>>>(h, b1, HID, nh, 1);

    // 3) mu = scatter(norm * (h@W_mu)) + b_mu   (written directly into d_out)
    k_gemm_f32    <<<(NN / 16) * (LAT / 16) / 8, 256, 0, stream>>>(h, Wmu, hu, NN, HID, LAT);
    k_selfinit    <<<nl / 256, 256, 0, stream>>>(hu, dinv, mu, LAT, nl);
    k_edge_scatter<<<NE / 8, 256, 0, stream>>>(hu, mu, srcI, dstI, dinv, LAT);
    k_bias_act    <<<nl / 256, 256, 0, stream>>>(mu, bmu, LAT, nl, 0);

    // 4) logvar = scatter(norm * (h@W_lv)) + b_lv (written directly into d_out)
    k_gemm_f32    <<<(NN / 16) * (LAT / 16) / 8, 256, 0, stream>>>(h, Wlv, hv, NN, HID, LAT);
    k_selfinit    <<<nl / 256, 256, 0, stream>>>(hv, dinv, lv, LAT, nl);
    k_edge_scatter<<<NE / 8, 256, 0, stream>>>(hv, lv, srcI, dstI, dinv, LAT);
    k_bias_act    <<<nl / 256, 256, 0, stream>>>(lv, blv, LAT, nl, 0);

    // 5) z = mu + eps * exp(0.5*logvar)
    k_reparam<<<nl / 256, 256, 0, stream>>>(mu, lv, eps, zb, nl);

    // 6) recon = sigmoid(z @ z^T)  — store-bandwidth bound (1 GiB out), WMMA f32
    const int reconWaves  = (NN / 32) * (NN / 32);      // 512*512
    k_recon<<<reconWaves / 8, 256, 0, stream>>>(zb, out);
}